// AttnResLayer_42915313221885
// MI455X (gfx1250) — compile-verified
//
#include <hip/hip_runtime.h>
#include <cstdint>

// Problem constants (from the reference): V[N,B,T,D] fp32
#define N_DEPTH 12
#define B_ 4
#define T_ 2048
#define D_ 1024
#define EPS_ 1e-6f
#define THREADS 256           // 8 wave32s
#define NWAVES (THREADS / 32)
#define VPT 4                 // D_ / THREADS floats per thread (one float4)
#define KCHUNK (D_ / (32 * 4))  // 8 float4 per lane when one wave sweeps a slice

typedef float v4f __attribute__((ext_vector_type(4)));
typedef int   v4i __attribute__((ext_vector_type(4)));
typedef __attribute__((address_space(1))) v4i g_v4i;   // global int4
typedef __attribute__((address_space(3))) v4i l_v4i;   // LDS int4

// ---- CDNA5 async global->LDS copy (ASYNCcnt-tracked), with safe fallback ----
#if defined(__has_builtin)
#  if __has_builtin(__builtin_amdgcn_global_load_async_to_lds_b128)
#    define HAVE_ASYNC_LDS 1
#  endif
#endif
#ifndef HAVE_ASYNC_LDS
#  define HAVE_ASYNC_LDS 0
#endif

__device__ __forceinline__ void async_copy_b128_nt(const float* gsrc, float* lds_dst) {
#if HAVE_ASYNC_LDS
    // Generic shared address: low 32 bits are the LDS byte offset (ISA aperture rule);
    // generic global address == addrspace(1) address numerically.
    // cpol = 1 -> TH_NT: V is streamed exactly once, keep it out of L2's way.
    __builtin_amdgcn_global_load_async_to_lds_b128(
        (g_v4i*)(uintptr_t)gsrc,
        (l_v4i*)(unsigned int)(uintptr_t)lds_dst,
        /*imm offset*/ 0, /*cpol TH_NT*/ 1);
#else
    *(v4f*)lds_dst = *(const v4f*)gsrc;
#endif
}

__device__ __forceinline__ void wait_async_copies() {
#if HAVE_ASYNC_LDS
#  if __has_builtin(__builtin_amdgcn_s_wait_asynccnt)
    __builtin_amdgcn_s_wait_asynccnt(0);
#  else
    asm volatile("s_wait_asynccnt 0x0" ::: "memory");
#  endif
#endif
}

// Fused: RMSNorm-stat + query dot + depth softmax + weighted sum, single HBM pass.
// One block per (b,t). V tile staged HBM->LDS via async DMA; each wave owns whole
// depth slices for the reductions (8x fewer shuffle ops than all-waves-reduce-all).
__global__ __launch_bounds__(THREADS)
void fused_depth_mix_kernel(const float* __restrict__ V,
                            const float* __restrict__ w_l,
                            const float* __restrict__ norm_w,
                            float* __restrict__ out) {
    __shared__ float sV[N_DEPTH * D_];              // 48 KB staging tile
    __shared__ __align__(16) float slog[N_DEPTH];   // depth logits

    const int tid  = threadIdx.x;
    const int lane = tid & 31;
    const int wave = tid >> 5;
    const int bt   = blockIdx.x;                    // flattened (b*T + t)
    const size_t baseBT = (size_t)bt * D_;
    const int d0 = tid * VPT;

    // Kick off all 12 depth-slice copies (global -> LDS, ASYNCcnt-tracked).
    const size_t n_stride = (size_t)B_ * T_ * D_;
#pragma unroll
    for (int n = 0; n < N_DEPTH; ++n) {
        async_copy_b128_nt(V + (size_t)n * n_stride + baseBT + d0,
                           &sV[n * D_ + d0]);
    }

    // Overlap with the async copies: per-lane query weights in the layout the
    // stats sweep uses (lane covers d = lane*4 + k*128). Cached, tiny.
    v4f wq[KCHUNK];
#pragma unroll
    for (int k = 0; k < KCHUNK; ++k) {
        const int d = lane * VPT + k * 128;
        v4f wl4 = *(const v4f*)(w_l    + d);
        v4f nw4 = *(const v4f*)(norm_w + d);
        wq[k] = wl4 * nw4;
    }

    wait_async_copies();
    __syncthreads();

    // Stats: wave w owns slices n = w, w+8. One shuffle chain per slice total.
    for (int n = wave; n < N_DEPTH; n += NWAVES) {
        float p2 = 0.0f, pw = 0.0f;
#pragma unroll
        for (int k = 0; k < KCHUNK; ++k) {
            const v4f x = *(const v4f*)&sV[n * D_ + lane * VPT + k * 128];
            p2 += x.x * x.x + x.y * x.y + x.z * x.z + x.w * x.w;
            pw += wq[k].x * x.x + wq[k].y * x.y + wq[k].z * x.z + wq[k].w * x.w;
        }
#pragma unroll
        for (int off = 16; off > 0; off >>= 1) {
            p2 += __shfl_down(p2, off, 32);
            pw += __shfl_down(pw, off, 32);
        }
        if (lane == 0) {
            const float rms = rsqrtf(p2 * (1.0f / (float)D_) + EPS_);
            slog[n] = pw * rms;   // logit_n = (Sum w*x) * rms
        }
    }
    __syncthreads();

    // Every thread: 12-wide softmax over depth (3 b128 LDS reads).
    float lg[N_DEPTH];
    {
        const v4f t0 = *(const v4f*)&slog[0];
        const v4f t1 = *(const v4f*)&slog[4];
        const v4f t2 = *(const v4f*)&slog[8];
        lg[0] = t0.x; lg[1]  = t0.y; lg[2]  = t0.z; lg[3]  = t0.w;
        lg[4] = t1.x; lg[5]  = t1.y; lg[6]  = t1.z; lg[7]  = t1.w;
        lg[8] = t2.x; lg[9]  = t2.y; lg[10] = t2.z; lg[11] = t2.w;
    }
    float mx = lg[0];
#pragma unroll
    for (int n = 1; n < N_DEPTH; ++n) mx = fmaxf(mx, lg[n]);
    float a[N_DEPTH];
    float denom = 0.0f;
#pragma unroll
    for (int n = 0; n < N_DEPTH; ++n) { a[n] = __expf(lg[n] - mx); denom += a[n]; }
    const float inv = __frcp_rn(denom);

    // Weighted sum over depth straight from the LDS tile; streamed NT store.
    v4f acc = (v4f)(0.0f);
#pragma unroll
    for (int n = 0; n < N_DEPTH; ++n) {
        const float an = a[n] * inv;
        const v4f x = *(const v4f*)&sV[n * D_ + d0];
        acc.x += an * x.x;
        acc.y += an * x.y;
        acc.z += an * x.z;
        acc.w += an * x.w;
    }
    __builtin_nontemporal_store(acc, (v4f*)(out + baseBT + d0));
}

extern "C" void kernel_launch(void* const* d_in, const int* in_sizes, int n_in,
                              void* d_out, int out_size, void* d_ws, size_t ws_size,
                              hipStream_t stream) {
    (void)in_sizes; (void)n_in; (void)d_ws; (void)ws_size; (void)out_size;
    const float* V      = (const float*)d_in[0];
    const float* w_l    = (const float*)d_in[1];
    const float* norm_w = (const float*)d_in[2];
    float* out          = (float*)d_out;

    dim3 grid(B_ * T_);   // one block per (b, t)
    dim3 block(THREADS);
    fused_depth_mix_kernel<<<grid, block, 0, stream>>>(V, w_l, norm_w, out);
}